// PointcloudGrouping_57123065037459
// MI455X (gfx1250) — compile-verified
//
#include <hip/hip_runtime.h>
#include <float.h>

typedef __attribute__((ext_vector_type(2))) float v2f;
typedef __attribute__((ext_vector_type(8))) float v8f;

#define NPTS   8192
#define NGRP   512
#define GSZ    32
#define NB     16
#define CHUNK  512
#define NCH    (NPTS / CHUNK)
#define CTILE  16

// ---------------- Kernel A: farthest point sampling (1 block / batch) --------
// Coordinates in registers (8 pts/thread). Per step: wave32 shuffle argmax,
// then one packed 64-bit LDS atomicMax ((f32bits(d)<<32)|~idx) for the
// cross-wave argmax -> 2 barriers per step, lowest-index tie-break preserved.
__global__ __launch_bounds__(1024) void fps_kernel(const float* __restrict__ pts,
                                                   float* __restrict__ centers)
{
  const int b    = blockIdx.x;
  const int tid  = threadIdx.x;
  const int lane = tid & 31;

  float xs[8], ys[8], zs[8], md[8];
#pragma unroll
  for (int i = 0; i < 8; ++i) {
    int n = tid + (i << 10);
    const float* p = pts + ((size_t)b * NPTS + n) * 6;
    xs[i] = p[0]; ys[i] = p[1]; zs[i] = p[2];
    md[i] = FLT_MAX;
  }

  __shared__ unsigned long long slots[2];
  __shared__ float s_pt[3];
  if (tid == 0) { slots[0] = 0ull; slots[1] = 0ull; }

  int cur = 0;
  for (int k = 0; k < NGRP; ++k) {
    // owner of `cur` publishes its coordinates and writes the center output
    if (tid == (cur & 1023)) {
      int slot = cur >> 10;
      s_pt[0] = xs[slot]; s_pt[1] = ys[slot]; s_pt[2] = zs[slot];
      float* c = centers + ((size_t)b * NGRP + k) * 3;
      c[0] = xs[slot]; c[1] = ys[slot]; c[2] = zs[slot];
    }
    __syncthreads();
    float lx = s_pt[0], ly = s_pt[1], lz = s_pt[2];

    float bv = -1.0f; int bi = 0;
#pragma unroll
    for (int i = 0; i < 8; ++i) {
      float dx = xs[i]-lx, dy = ys[i]-ly, dz = zs[i]-lz;
      float d = dx*dx + dy*dy + dz*dz;
      md[i] = fminf(md[i], d);
      if (md[i] > bv) { bv = md[i]; bi = tid + (i << 10); }
    }
    // wave argmax, lowest index wins ties (matches jnp.argmax)
#pragma unroll
    for (int off = 16; off >= 1; off >>= 1) {
      float ov = __shfl_xor(bv, off, 32);
      int   oi = __shfl_xor(bi, off, 32);
      if (ov > bv || (ov == bv && oi < bi)) { bv = ov; bi = oi; }
    }
    if (lane == 0) {
      unsigned long long key =
          ((unsigned long long)__float_as_uint(bv) << 32) |
          (unsigned int)(~bi);
      atomicMax(&slots[k & 1], key);
    }
    if (tid == 0) slots[(k + 1) & 1] = 0ull;   // reset slot for next step
    __syncthreads();
    cur = (int)(~(unsigned int)slots[k & 1]);
  }
}

// ------------- Kernel B: WMMA distance tiles + wave bitonic top-32 -----------
// Block = (batch, 16-center tile); 16 waves. d2 = A(-2c, K=4 padded) x B(xyz)
// + C(||c||^2 + ||x||^2) via V_WMMA_F32_16X16X4_F32. Chunk xyz is double-
// buffered in LDS and prefetched with GLOBAL_LOAD_ASYNC_TO_LDS_B32 (ASYNCcnt).
// Each wave owns one center and keeps its best-32 sorted across lanes.
__device__ __forceinline__ void prefetch_chunk(const float* __restrict__ pts,
                                               int b, int base, int tid,
                                               float* dpx, float* dpy, float* dpz)
{
  unsigned long long ga =
      (unsigned long long)(uintptr_t)(pts + ((size_t)b * NPTS + base + tid) * 6);
  unsigned lx = (unsigned)(uintptr_t)(dpx + tid);
  unsigned ly = (unsigned)(uintptr_t)(dpy + tid);
  unsigned lz = (unsigned)(uintptr_t)(dpz + tid);
  // INST_OFFSET applies to both LDS and global addresses on async ops, so the
  // +4/+8 is baked into the global address instead of using offset:.
  asm volatile("global_load_async_to_lds_b32 %0, %1, off"
               :: "v"(lx), "v"(ga) : "memory");
  asm volatile("global_load_async_to_lds_b32 %0, %1, off"
               :: "v"(ly), "v"(ga + 4ull) : "memory");
  asm volatile("global_load_async_to_lds_b32 %0, %1, off"
               :: "v"(lz), "v"(ga + 8ull) : "memory");
}

__global__ __launch_bounds__(512) void group_kernel(const float* __restrict__ pts,
                                                    const float* __restrict__ centers,
                                                    float* __restrict__ grouped)
{
  const int b    = blockIdx.x >> 5;
  const int g0   = (blockIdx.x & 31) * CTILE;
  const int tid  = threadIdx.x;
  const int lane = tid & 31;
  const int w    = tid >> 5;
  const bool hi  = lane >= 16;
  const int  ml  = lane & 15;

  __shared__ float cxs[CTILE], cys[CTILE], czs[CTILE], cn2[CTILE];
  __shared__ float px[2][CHUNK], py[2][CHUNK], pz[2][CHUNK];
  __shared__ float d2buf[CTILE * CHUNK];

  // kick off chunk 0 while we fetch the centers
  prefetch_chunk(pts, b, 0, tid, px[0], py[0], pz[0]);

  if (tid < CTILE) {
    const float* c = centers + ((size_t)b * NGRP + g0 + tid) * 3;
    float X = c[0], Y = c[1], Z = c[2];
    cxs[tid] = X; cys[tid] = Y; czs[tid] = Z;
    cn2[tid] = X*X + Y*Y + Z*Z;
  }
  __syncthreads();

  // A fragment (16x4 f32): lanes 0-15 -> K0,K1 ; lanes 16-31 -> K2,K3(pad)
  v2f av;
  av.x = hi ? -2.f * czs[ml] : -2.f * cxs[ml];
  av.y = hi ? 0.f            : -2.f * cys[ml];

  float bvv = FLT_MAX;          // best-32 values, sorted ascending across lanes
  int   bii = 0x7fffffff;       // matching point indices

  for (int c = 0; c < NCH; ++c) {
    const int buf  = c & 1;
    const int base = c * CHUNK;

    if (c + 1 < NCH) {
      prefetch_chunk(pts, b, base + CHUNK, tid, px[buf ^ 1], py[buf ^ 1], pz[buf ^ 1]);
      // async ops complete in order: <=3 outstanding => chunk c's 3 are done
      asm volatile("s_wait_asynccnt 0x3" ::: "memory");
    } else {
      asm volatile("s_wait_asynccnt 0x0" ::: "memory");
    }
    __syncthreads();   // chunk c staged everywhere; select(c-1) fully drained

    // each wave computes two 16x16 d2 tiles
#pragma unroll
    for (int t = 0; t < 2; ++t) {
      int pt0 = (w * 2 + t) * 16;
      int nl  = pt0 + ml;
      float X = px[buf][nl], Y = py[buf][nl], Z = pz[buf][nl];
      float p2 = X*X + Y*Y + Z*Z;
      v2f bf;
      bf.x = hi ? Z   : X;    // K2 : K0 (value select, no exec divergence)
      bf.y = hi ? 0.f : Y;    // K3 : K1
      int mb = hi ? 8 : 0;
      v8f cf;
#pragma unroll
      for (int r = 0; r < 8; ++r) cf[r] = cn2[mb + r] + p2;
      v8f df = __builtin_amdgcn_wmma_f32_16x16x4_f32(false, av, false, bf,
                                                     (short)0, cf, false, false);
#pragma unroll
      for (int r = 0; r < 8; ++r) d2buf[(mb + r) * CHUNK + nl] = df[r];
    }
    __syncthreads();

    // selection: wave w owns center g0+w; exact top-32 via bitonic merges
    float thr = __shfl(bvv, 31, 32);
    int   ti  = __shfl(bii, 31, 32);
#pragma unroll 1
    for (int jj = 0; jj < CHUNK / 32; ++jj) {
      int   nloc = jj * 32 + lane;
      float cv = d2buf[w * CHUNK + nloc];
      int   ci = base + nloc;
      bool better = (cv < thr) || (cv == thr && ci < ti);
      if (__any(better)) {                 // wave-uniform; EXEC stays full
        // bitonic sort the 32 candidates ascending (lex on (val, idx))
        for (int kk = 2; kk <= 32; kk <<= 1) {
          for (int j = kk >> 1; j > 0; j >>= 1) {
            float ov = __shfl_xor(cv, j, 32);
            int   oi = __shfl_xor(ci, j, 32);
            bool asc  = (lane & kk) == 0;
            bool low  = (lane & j)  == 0;
            bool less = (cv < ov) || (cv == ov && ci < oi);
            if (((asc == low) != less)) { cv = ov; ci = oi; }
          }
        }
        // keep 32 smallest of (best asc, cand asc): L[i]=min(A[i],B[31-i])
        float rv = __shfl_xor(cv, 31, 32);
        int   ri = __shfl_xor(ci, 31, 32);
        bool less = (rv < bvv) || (rv == bvv && ri < bii);
        if (less) { bvv = rv; bii = ri; }
        // clean bitonic sequence back to ascending
        for (int j = 16; j > 0; j >>= 1) {
          float ov = __shfl_xor(bvv, j, 32);
          int   oi = __shfl_xor(bii, j, 32);
          bool low   = (lane & j) == 0;
          bool less2 = (bvv < ov) || (bvv == ov && bii < oi);
          if (low != less2) { bvv = ov; bii = oi; }
        }
        thr = __shfl(bvv, 31, 32);
        ti  = __shfl(bii, 31, 32);
      }
    }
    // no trailing barrier needed: next iteration's post-wait barrier orders
    // select(c) reads of d2buf before compute(c+1) overwrites it
  }

  // gather + center-subtract; lane l writes neighbor l (ascending d2 order)
  const int g = g0 + w;
  const float* p = pts + ((size_t)b * NPTS + bii) * 6;
  float* o = grouped + (((size_t)(b * NGRP + g)) * GSZ + lane) * 6;
  o[0] = p[0] - cxs[w];
  o[1] = p[1] - cys[w];
  o[2] = p[2] - czs[w];
  o[3] = p[3];
  o[4] = p[4];
  o[5] = p[5];
}

extern "C" void kernel_launch(void* const* d_in, const int* in_sizes, int n_in,
                              void* d_out, int out_size, void* d_ws, size_t ws_size,
                              hipStream_t stream) {
  const float* pts = (const float*)d_in[0];
  float* grouped = (float*)d_out;
  float* centers = grouped + (size_t)NB * NGRP * GSZ * 6;  // output tuple order

  fps_kernel<<<NB, 1024, 0, stream>>>(pts, centers);
  group_kernel<<<NB * (NGRP / CTILE), 512, 0, stream>>>(pts, centers, grouped);
}